// GraphQuantumNeuralNetwork_65481071395343
// MI455X (gfx1250) — compile-verified
//
#include <hip/hip_runtime.h>

#define NQ     15
#define DIM    (1 << NQ)        // 32768 amplitudes (256 KB complex64) -> fits one WGP's LDS
#define DEPTH  4
#define TA     1024             // 32 waves on one WGP
#define TB     256
#define PERT   (DIM / TA)       // 32 amplitudes per thread in state kernel
#define NTILES (DIM / 16 / 16)  // 128 column-tiles per 4-qubit group pass

typedef float v2f __attribute__((ext_vector_type(2)));
typedef float v8f __attribute__((ext_vector_type(8)));

struct cplx { float x, y; };
__device__ __forceinline__ cplx cmul(cplx a, cplx b) {
    return { __fmaf_rn(a.x, b.x, -a.y * b.y), __fmaf_rn(a.x, b.y, a.y * b.x) };
}

// Fused U = Rz(t2)*Ry(t1)*Rx(t0) (applied RX first, as in the reference).
__device__ __forceinline__ void build_gate(const float* __restrict__ tq, cplx m[2][2]) {
    float sx, cx, sy, cy, sz, cz;
    sincosf(0.5f * tq[0], &sx, &cx);
    sincosf(0.5f * tq[1], &sy, &cy);
    sincosf(0.5f * tq[2], &sz, &cz);
    cplx A00 = {  cy * cx,  sy * sx };
    cplx A01 = { -sy * cx, -cy * sx };
    cplx A10 = {  sy * cx, -cy * sx };
    cplx A11 = {  cy * cx, -sy * sx };
    cplx z0 = { cz, -sz }, z1 = { cz, sz };
    m[0][0] = cmul(z0, A00); m[0][1] = cmul(z0, A01);
    m[1][0] = cmul(z1, A10); m[1][1] = cmul(z1, A11);
}

// Inverse of the CNOT ring CNOT(0,1)..CNOT(13,14),CNOT(14,0); wire 0 = MSB.
__device__ __forceinline__ int cnot_ring_src(int y) {
    int x = y ^ ((y & 1) << (NQ - 1));
    #pragma unroll
    for (int c = NQ - 2; c >= 0; --c)
        x ^= (((x >> (NQ - 1 - c)) & 1) << (NQ - 2 - c));
    return x;
}

// ---------------------------------------------------------------------------
// Kernel A: state resident in 256 KB LDS; 4-qubit gate groups applied as
// 16x16 complex matmuls on the matrix unit (v_wmma_f32_16x16x4_f32), CNOT
// ring as one register-staged permutation, async LDS->global spill of V.
// ---------------------------------------------------------------------------
__global__ __launch_bounds__(TA, 1)
void qsim_state_kernel(const float* __restrict__ theta, float2* __restrict__ vout) {
    extern __shared__ float2 st[];
    const int tid  = threadIdx.x;
    const int lane = tid & 31;
    const int wave = tid >> 5;
    const int row  = lane & 15;     // M / N index within a 16-wide tile
    const int half = lane >> 4;     // lane group 0..1

    for (int i = tid; i < DIM; i += TA) st[i] = make_float2(0.f, 0.f);
    if (tid == 0) st[0] = make_float2(1.f, 0.f);
    __syncthreads();

    for (int d = 0; d < DEPTH; ++d) {
        // --- four 4-qubit gate groups: fields at bit positions p..p+3 ---
        for (int g = 0; g < 4; ++g) {
            const int p = (g < 3) ? (11 - 4 * g) : 0;   // 11,7,3,0

            // Per-lane A operands: U[r][s] for s = 4t + 2*half + v (ISA A layout).
            v2f Ar[4], Ai[4];
            {
                cplx m[4][2][2];
                #pragma unroll
                for (int j = 0; j < 4; ++j) {
                    if (g == 3 && j == 3) {             // identity pad (qubit 11 done in g=2)
                        m[j][0][0] = {1.f, 0.f}; m[j][0][1] = {0.f, 0.f};
                        m[j][1][0] = {0.f, 0.f}; m[j][1][1] = {1.f, 0.f};
                    } else {
                        const int q = 14 - p - j;       // qubit owning field bit j
                        build_gate(theta + ((d * NQ) + q) * 3, m[j]);
                    }
                }
                #pragma unroll
                for (int t = 0; t < 4; ++t) {
                    #pragma unroll
                    for (int v = 0; v < 2; ++v) {
                        const int s = 4 * t + 2 * half + v;
                        cplx u = cmul(cmul(m[0][row & 1][s & 1],
                                           m[1][(row >> 1) & 1][(s >> 1) & 1]),
                                      cmul(m[2][(row >> 2) & 1][(s >> 2) & 1],
                                           m[3][(row >> 3) & 1][(s >> 3) & 1]));
                        Ar[t][v] = u.x; Ai[t][v] = u.y;
                    }
                }
            }

            const int lowm = (1 << p) - 1;
            for (int tl = wave; tl < NTILES; tl += TA / 32) {
                const int c    = tl * 16 + row;                       // this lane's column
                const int base = ((c >> p) << (p + 4)) | (c & lowm);  // field bits cleared

                // B operands: X[k][c], k = 4t + 2*half + v (ISA B layout).
                v2f Br[4], Bi[4], Bin[4];
                #pragma unroll
                for (int t = 0; t < 4; ++t) {
                    const int k0 = 4 * t + 2 * half;
                    float2 w0 = st[base | (k0 << p)];
                    float2 w1 = st[base | ((k0 + 1) << p)];
                    Br[t][0] = w0.x;  Br[t][1] = w1.x;
                    Bi[t][0] = w0.y;  Bi[t][1] = w1.y;
                    Bin[t][0] = -w0.y; Bin[t][1] = -w1.y;
                }

                // Yr = Ur*Xr + Ui*(-Xi);  Yi = Ur*Xi + Ui*Xr   (K=16 via 4 slices)
                v8f yr = {0.f,0.f,0.f,0.f,0.f,0.f,0.f,0.f};
                v8f yi = {0.f,0.f,0.f,0.f,0.f,0.f,0.f,0.f};
                #pragma unroll
                for (int t = 0; t < 4; ++t)
                    yr = __builtin_amdgcn_wmma_f32_16x16x4_f32(false, Ar[t], false, Br[t],
                                                               (short)0, yr, false, false);
                #pragma unroll
                for (int t = 0; t < 4; ++t)
                    yr = __builtin_amdgcn_wmma_f32_16x16x4_f32(false, Ai[t], false, Bin[t],
                                                               (short)0, yr, false, false);
                #pragma unroll
                for (int t = 0; t < 4; ++t)
                    yi = __builtin_amdgcn_wmma_f32_16x16x4_f32(false, Ar[t], false, Bi[t],
                                                               (short)0, yi, false, false);
                #pragma unroll
                for (int t = 0; t < 4; ++t)
                    yi = __builtin_amdgcn_wmma_f32_16x16x4_f32(false, Ai[t], false, Br[t],
                                                               (short)0, yi, false, false);

                // D layout: VGPR v holds output row M = v + 8*half at column c.
                #pragma unroll
                for (int v = 0; v < 8; ++v) {
                    const int rr = v + 8 * half;
                    st[base | (rr << p)] = make_float2(yr[v], yi[v]);
                }
            }
            __syncthreads();
        }

        // --- entire CNOT ring as ONE register-staged gather permutation ---
        float2 tmp[PERT];
        #pragma unroll
        for (int j = 0; j < PERT; ++j)
            tmp[j] = st[cnot_ring_src(tid + j * TA)];
        __syncthreads();
        #pragma unroll
        for (int j = 0; j < PERT; ++j)
            st[tid + j * TA] = tmp[j];
        __syncthreads();
    }

    // Async DMA spill: LDS -> global (GLOBAL_STORE_ASYNC_FROM_LDS_B128).
    {
        unsigned long long gaddr = (unsigned long long)(vout + (size_t)tid * PERT);
        unsigned int       laddr = (unsigned int)(unsigned long long)&st[tid * PERT];
        #pragma unroll
        for (int k = 0; k < (PERT * 8) / 16; ++k) {
            asm volatile("global_store_async_from_lds_b128 %0, %1, off offset:%2"
                         :: "v"(gaddr), "v"(laddr), "i"(16 * k) : "memory");
        }
        asm volatile("s_wait_asynccnt 0" ::: "memory");
    }
}

// ---------------------------------------------------------------------------
// Kernel B: out[b,:] = e^{-i * 0.5*sum_q inputs[b,q]} * V[:]   (64 MB stream)
// ---------------------------------------------------------------------------
__global__ __launch_bounds__(TB, 1)
void qsim_broadcast_kernel(const float* __restrict__ inputs,
                           const float4* __restrict__ v,
                           float4* __restrict__ out) {
    const int b   = blockIdx.x;
    const int tid = threadIdx.x;

    float S = 0.f;
    #pragma unroll
    for (int q = 0; q < NQ; ++q) S += inputs[b * NQ + q];
    S *= 0.5f;
    float sS, cS;
    sincosf(S, &sS, &cS);                              // phase = cS - i*sS

    float4* orow = out + (size_t)b * (DIM / 2);
    for (int k = tid; k < DIM / 2; k += TB) {
        float4 w = v[k];
        float4 r;
        r.x = __fmaf_rn(cS, w.x,  sS * w.y);
        r.y = __fmaf_rn(cS, w.y, -sS * w.x);
        r.z = __fmaf_rn(cS, w.z,  sS * w.w);
        r.w = __fmaf_rn(cS, w.w, -sS * w.z);
        orow[k] = r;
    }
}

extern "C" void kernel_launch(void* const* d_in, const int* in_sizes, int n_in,
                              void* d_out, int out_size, void* d_ws, size_t ws_size,
                              hipStream_t stream) {
    const float* inputs = (const float*)d_in[0];       // [B, 15]
    const float* theta  = (const float*)d_in[1];       // [4, 15, 3]
    const int B = in_sizes[0] / NQ;                    // 256

    float2* vbuf = (float2*)d_ws;                      // 256 KB scratch

    qsim_state_kernel<<<1, TA, DIM * sizeof(float2), stream>>>(theta, vbuf);
    qsim_broadcast_kernel<<<B, TB, 0, stream>>>(inputs, (const float4*)vbuf,
                                                (float4*)d_out);
}